// FuzzyDirGCNConv_77773267796194
// MI455X (gfx1250) — compile-verified
//
#include <hip/hip_runtime.h>

#define N_NODES 100000
#define N_EDGES 1250000
#define F       64

typedef __attribute__((ext_vector_type(2))) float v2f;
typedef __attribute__((ext_vector_type(8))) float v8f;

// ---------------------------------------------------------------------------
// Zero-init the two aggregation buffers in workspace (graph-capture safe).
// ---------------------------------------------------------------------------
__global__ __launch_bounds__(256) void zero_kernel(float* __restrict__ p, int n) {
    int i = blockIdx.x * blockDim.x + threadIdx.x;
    if (i < n) p[i] = 0.0f;
}

// ---------------------------------------------------------------------------
// Edge scatter: one wave32 per edge. Lane L handles features [2L, 2L+1].
// x-row gather is a fully coalesced 256B wave read; scatter adds are
// global_atomic_add_f32 (no return) landing in the 192MB L2 (agg buffers are
// only 51.2MB total, L2-resident on MI455X).
// ---------------------------------------------------------------------------
__global__ __launch_bounds__(256) void scatter_kernel(
    const float* __restrict__ x,
    const int*   __restrict__ edge_index,
    const float* __restrict__ edge_weight,
    float* __restrict__ agg_s2d,
    float* __restrict__ agg_d2s) {

    const int lane = threadIdx.x & 31;
    const int e    = (blockIdx.x * blockDim.x + threadIdx.x) >> 5;
    if (e >= N_EDGES) return;

    const int   s  = edge_index[e];            // sender
    const int   r  = edge_index[N_EDGES + e];  // receiver
    const float ws = edge_weight[e];           // edge_weight[0, e, 0]
    const float wd = edge_weight[N_EDGES + e]; // edge_weight[1, e, 0]

    const float2 xv = *(const float2*)(x + (size_t)s * F + lane * 2);

    float* ps = agg_s2d + (size_t)r * F + lane * 2;
    float* pd = agg_d2s + (size_t)r * F + lane * 2;
    atomicAdd(ps + 0, xv.x * ws);
    atomicAdd(ps + 1, xv.y * ws);
    atomicAdd(pd + 0, xv.x * wd);
    atomicAdd(pd + 1, xv.y * wd);
}

// ---------------------------------------------------------------------------
// Node GEMM via V_WMMA_F32_16X16X4_F32 (exact fp32, matches reference dtype).
// One wave32 computes one 16x16 output tile: 16 WMMAs over K=64.
// wg encodes {node tile (6250), matrix sel (2), col tile (4)} -> 50000 waves.
// VGPR layouts per CDNA5 ISA 7.12.2:
//   A 16x4 f32 : lane l, group g -> A[l][k+2g], A[l][k+2g+1]
//   B 4x16 f32 : lane l, group g -> B[k+2g][n], B[k+2g+1][n]
//   C/D 16x16  : vgpr r, lane l, group g -> D[r + 8g][n]
// ---------------------------------------------------------------------------
__global__ __launch_bounds__(256) void gemm_wmma_kernel(
    const float* __restrict__ agg_s2d,
    const float* __restrict__ agg_d2s,
    const float* __restrict__ W_src,
    const float* __restrict__ W_dst,
    const float* __restrict__ b_src,
    const float* __restrict__ b_dst,
    float* __restrict__ out) {

    const int lane = threadIdx.x & 31;
    const int wg   = blockIdx.x * 8 + (threadIdx.x >> 5);  // global wave id: 0..49999

    const int nTile = wg & 3;          // 16-col tile within F_OUT=64
    const int mat   = (wg >> 2) & 1;   // 0 = s2d/W_src, 1 = d2s/W_dst
    const int mTile = wg >> 3;         // node tile: 0..6249

    const float* __restrict__ A    = mat ? agg_d2s : agg_s2d;
    const float* __restrict__ W    = mat ? W_dst   : W_src;
    const float* __restrict__ bias = mat ? b_dst   : b_src;
    float* __restrict__ O = out + (size_t)mat * N_NODES * F;

    const int g   = lane >> 4;   // lane group (0: lanes 0-15, 1: lanes 16-31)
    const int l   = lane & 15;
    const int row = mTile * 16 + l;
    const int n0  = nTile * 16;

    v8f c = {};
#pragma unroll
    for (int k = 0; k < F; k += 4) {
        // A fragment (contiguous float2 per lane)
        const float2 av = *(const float2*)(A + (size_t)row * F + k + 2 * g);
        v2f a; a.x = av.x; a.y = av.y;
        // B fragment (two coalesced row segments of W)
        v2f b;
        b.x = W[(k + 2 * g)     * F + n0 + l];
        b.y = W[(k + 2 * g + 1) * F + n0 + l];
        // D = A x B + C  (fp32 WMMA, round-to-nearest-even)
        c = __builtin_amdgcn_wmma_f32_16x16x4_f32(
                /*neg_a=*/false, a, /*neg_b=*/false, b,
                /*c_mod=*/(short)0, c, /*reuse_a=*/false, /*reuse_b=*/false);
    }

    const float bv = bias[n0 + l];
#pragma unroll
    for (int r = 0; r < 8; ++r) {
        O[(size_t)(mTile * 16 + r + 8 * g) * F + n0 + l] = c[r] + bv;
    }
}

// ---------------------------------------------------------------------------
// Launcher: inputs in setup_inputs() order:
//   0: x [N,64] f32        1: edge_index [2,E] i32   2: edge_weight [2,E,1] f32
//   3: W_src [64,64] f32   4: W_dst [64,64] f32
//   5: b_src [64] f32      6: b_dst [64] f32
// d_out: (out_s2d, out_d2s) concatenated -> 2*N*64 f32.
// d_ws : 2*N*64 f32 aggregation scratch (51.2 MB).
// ---------------------------------------------------------------------------
extern "C" void kernel_launch(void* const* d_in, const int* in_sizes, int n_in,
                              void* d_out, int out_size, void* d_ws, size_t ws_size,
                              hipStream_t stream) {
    const float* x           = (const float*)d_in[0];
    const int*   edge_index  = (const int*)  d_in[1];
    const float* edge_weight = (const float*)d_in[2];
    const float* W_src       = (const float*)d_in[3];
    const float* W_dst       = (const float*)d_in[4];
    const float* b_src       = (const float*)d_in[5];
    const float* b_dst       = (const float*)d_in[6];
    float* out = (float*)d_out;

    float* agg_s2d = (float*)d_ws;
    float* agg_d2s = agg_s2d + (size_t)N_NODES * F;

    const int totalAgg = 2 * N_NODES * F;
    zero_kernel<<<(totalAgg + 255) / 256, 256, 0, stream>>>(agg_s2d, totalAgg);

    const int scatterBlocks = (N_EDGES * 32) / 256;  // 1 wave per edge, 8 waves/block
    scatter_kernel<<<scatterBlocks, 256, 0, stream>>>(
        x, edge_index, edge_weight, agg_s2d, agg_d2s);

    // 6250 node tiles x (4 col tiles x 2 matrices) waves / 8 waves per block
    gemm_wmma_kernel<<<N_NODES / 16, 256, 0, stream>>>(
        agg_s2d, agg_d2s, W_src, W_dst, b_src, b_dst, out);
}